// CrossAttention_69269232550341
// MI455X (gfx1250) — compile-verified
//
#include <hip/hip_runtime.h>

typedef float v2f __attribute__((ext_vector_type(2)));
typedef float v8f __attribute__((ext_vector_type(8)));

#define WMMA_F32(a, b, c) \
  __builtin_amdgcn_wmma_f32_16x16x4_f32(false, (a), false, (b), (short)0, (c), false, false)

// ---------------------------------------------------------------------------
// CDNA5 async global->LDS copy (16 bytes per lane, ASYNCcnt-tracked).
// per-lane LDS address comes from a VGPR; generic shared pointer's low 32 bits
// are the LDS byte offset (aperture truncation).
// ---------------------------------------------------------------------------
__device__ __forceinline__ void async_ld16(void* lds_dst, const void* gsrc) {
    unsigned loff = (unsigned)(unsigned long long)lds_dst;
    asm volatile("global_load_async_to_lds_b128 %0, %1, off"
                 :: "v"(loff), "v"(gsrc)
                 : "memory");
}
__device__ __forceinline__ void wait_async0() {
    asm volatile("s_wait_asynccnt 0x0" ::: "memory");
}

// ---------------------------------------------------------------------------
// GEMM + bias:  C[M,N] = A[M,K] @ W[K,N] + bias[N]
// 256 thr = 8 waves, 64x64 block tile, wave = 16x32 slab (2 wmma accums).
// A tile row-major (stride 68, async-staged); W tile k-pair interleaved
// (v2f [k/2][n], row stride 80 v2f) so each B fragment is one ds_load_b64.
// ---------------------------------------------------------------------------
__global__ __launch_bounds__(256)
void gemm_bias_kernel(const float* __restrict__ A, const float* __restrict__ W,
                      const float* __restrict__ bias, float* __restrict__ C,
                      int M, int N, int K)
{
    __shared__ float sA[64][68];
    __shared__ v2f   sWp[32 * 80];     // (k/2) * 80 + n

    const int tid  = threadIdx.x;
    const int wv   = tid >> 5;
    const int lane = tid & 31;
    const int lm   = lane & 15;
    const int hi   = lane >> 4;
    const int m0   = blockIdx.x * 64;
    const int n0   = blockIdx.y * 64;
    const int wm   = wv >> 1;
    const int wn   = wv & 1;

    v8f acc0 = {0.f,0.f,0.f,0.f,0.f,0.f,0.f,0.f};
    v8f acc1 = {0.f,0.f,0.f,0.f,0.f,0.f,0.f,0.f};

    for (int kc = 0; kc < K; kc += 64) {
#pragma unroll
        for (int i = 0; i < 4; ++i) {
            int slot = tid + i * 256;          // 1024 float4 slots per tile
            int r  = slot >> 4;
            int c4 = (slot & 15) * 4;
            int gr = m0 + r; if (gr >= M) gr = M - 1;   // clamp; store is guarded
            async_ld16(&sA[r][c4], A + (size_t)gr * K + kc + c4);
            float4 vw = *(const float4*)(W + (size_t)(kc + r) * N + n0 + c4);
            float* wd = (float*)sWp + (r >> 1) * 160 + (r & 1);
            wd[(c4 + 0) * 2] = vw.x; wd[(c4 + 1) * 2] = vw.y;
            wd[(c4 + 2) * 2] = vw.z; wd[(c4 + 3) * 2] = vw.w;
        }
        wait_async0();
        __syncthreads();

#pragma unroll
        for (int ks = 0; ks < 16; ++ks) {
            int kb = ks * 4 + 2 * hi;          // even k of this lane-half's pair
            v2f a  = *(const v2f*)&sA[wm * 16 + lm][kb];
            v2f b0 = sWp[(kb >> 1) * 80 + wn * 32 + lm];
            v2f b1 = sWp[(kb >> 1) * 80 + wn * 32 + 16 + lm];
            acc0 = WMMA_F32(a, b0, acc0);
            acc1 = WMMA_F32(a, b1, acc1);
        }
        __syncthreads();
    }

#pragma unroll
    for (int r = 0; r < 8; ++r) {
        int gm = m0 + wm * 16 + r + 8 * hi;    // D: lanes>=16 hold M=r+8
        if (gm < M) {
            int c0 = n0 + wn * 32 + lm;
            C[(size_t)gm * N + c0]      = acc0[r] + bias[c0];
            C[(size_t)gm * N + c0 + 16] = acc1[r] + bias[c0 + 16];
        }
    }
}

// ---------------------------------------------------------------------------
// Fused attention per (q-tile 16, head, batch):
//   S = (Q Kh^T)/8 + CPB-MLP(rel)  -> row softmax -> O = P Vh
// ---------------------------------------------------------------------------
#define SST 1028   // padded S row stride
#define VST 68     // K-stage row stride (row-major phase)
#define KVSTAGE 20480  // floats: max(256*68, 128*160)

__global__ __launch_bounds__(256)
void attn_kernel(const float* __restrict__ q,   // (4,600,512)
                 const float* __restrict__ kv,  // (4,1024,8*128): +0..63 K, +64..127 V
                 const float* __restrict__ rel, // (4,600,1024,2)
                 const float* __restrict__ w1, const float* __restrict__ b1,
                 const float* __restrict__ w2, const float* __restrict__ b2,
                 float* __restrict__ ctx)       // (4,600,512)
{
    extern __shared__ float smem[];
    float* S    = smem;               // 16 * SST
    float* Kst  = S + 16 * SST;       // KVSTAGE (K chunks row-major / V pairs)
    float* w1s  = Kst + KVSTAGE;      // 256
    float* b1s  = w1s + 256;          // 128
    float* w2s  = b1s + 128;          // 128 (column h of cpb_w2)
    float* Ored = w2s + 128;          // 1024 split-K combine

    const int tid  = threadIdx.x;
    const int wv   = tid >> 5;
    const int lane = tid & 31;
    const int lm   = lane & 15;
    const int hi   = lane >> 4;
    const int qt   = blockIdx.x;
    const int h    = blockIdx.y;
    const int b    = blockIdx.z;
    const int q0   = qt * 16;

    w1s[tid] = w1[tid];
    if (tid < 128) { b1s[tid] = b1[tid]; w2s[tid] = w2[tid * 8 + h]; }
    const float b2h = b2[h];

    // ---- preload Q fragments: all 16 k-steps over d=64, one v2f each
    int qr = q0 + lm; if (qr > 599) qr = 599;
    const float* qp = q + (size_t)(b * 600 + qr) * 512 + h * 64;
    v2f afr[16];
#pragma unroll
    for (int ks = 0; ks < 16; ++ks)
        afr[ks] = *(const v2f*)(qp + ks * 4 + 2 * hi);

    // ---- Phase 1: S = Q K^T  (K streamed in 256-row chunks, async->LDS)
    for (int c = 0; c < 4; ++c) {
        int kc0 = c * 256;
#pragma unroll
        for (int i = 0; i < 16; ++i) {
            int slot = tid + i * 256;           // 4096 float4 slots = 256x64
            int r  = slot >> 4;
            int c4 = (slot & 15) * 4;
            async_ld16(Kst + r * VST + c4,
                       kv + (size_t)(b * 1024 + kc0 + r) * 1024 + h * 128 + c4);
        }
        wait_async0();
        __syncthreads();

#pragma unroll
        for (int tt = 0; tt < 2; ++tt) {
            int kb = (wv * 2 + tt) * 16;
            v8f acc = {0.f,0.f,0.f,0.f,0.f,0.f,0.f,0.f};
#pragma unroll
            for (int ks = 0; ks < 16; ++ks) {
                int kk = ks * 4 + 2 * hi;
                v2f bb = *(const v2f*)&Kst[(kb + lm) * VST + kk];
                acc = WMMA_F32(afr[ks], bb, acc);
            }
#pragma unroll
            for (int r = 0; r < 8; ++r)
                S[(r + 8 * hi) * SST + kc0 + kb + lm] = acc[r];
        }
        __syncthreads();
    }

    // ---- Phase 2: fused CPB MLP + 1/sqrt(64) scale, in place on S
    for (int p = 0; p < 64; ++p) {
        int flat = tid + p * 256;
        int row  = flat >> 10;
        int col  = flat & 1023;
        int qq   = q0 + row; if (qq > 599) qq = 599;
        v2f d = *(const v2f*)(rel + ((size_t)(b * 600 + qq) * 1024 + col) * 2);
        float acc = b2h;
#pragma unroll 8
        for (int j = 0; j < 128; ++j) {
            float hv = fmaf(d.x, w1s[j], fmaf(d.y, w1s[128 + j], b1s[j]));
            hv  = fmaxf(hv, 0.0f);
            acc = fmaf(hv, w2s[j], acc);
        }
        float* sp = S + row * SST + col;
        *sp = *sp * 0.125f + acc;
    }
    __syncthreads();

    // ---- Phase 3: row softmax (wave owns 2 rows; wave32 shuffles)
#pragma unroll
    for (int rr = 0; rr < 2; ++rr) {
        int row = wv * 2 + rr;
        float* sr = S + row * SST;
        float ev[32];
        float mx = -3.0e38f;
#pragma unroll
        for (int i = 0; i < 32; ++i) { ev[i] = sr[lane + i * 32]; mx = fmaxf(mx, ev[i]); }
#pragma unroll
        for (int off = 16; off > 0; off >>= 1) mx = fmaxf(mx, __shfl_xor(mx, off, 32));
        float sum = 0.0f;
#pragma unroll
        for (int i = 0; i < 32; ++i) { ev[i] = __expf(ev[i] - mx); sum += ev[i]; }
#pragma unroll
        for (int off = 16; off > 0; off >>= 1) sum += __shfl_xor(sum, off, 32);
        float inv = 1.0f / sum;
#pragma unroll
        for (int i = 0; i < 32; ++i) sr[lane + i * 32] = ev[i] * inv;
    }
    __syncthreads();

    // ---- Phase 4: O = P V ; V staged k-pair interleaved so B frags are b64
    const int dt    = wv & 3;
    const int half  = wv >> 2;
    const int dbase = dt * 16;
    v2f* Vp = (v2f*)Kst;               // (k/2)*80 + d
    v8f acc = {0.f,0.f,0.f,0.f,0.f,0.f,0.f,0.f};

    for (int c = 0; c < 4; ++c) {
        int kc0 = c * 256;
#pragma unroll
        for (int i = 0; i < 16; ++i) {
            int slot = tid + i * 256;
            int r  = slot >> 4;
            int c4 = (slot & 15) * 4;
            float4 vv = *(const float4*)(kv + (size_t)(b * 1024 + kc0 + r) * 1024
                                            + h * 128 + 64 + c4);
            float* dst = Kst + (r >> 1) * 160 + (r & 1);
            dst[(c4 + 0) * 2] = vv.x; dst[(c4 + 1) * 2] = vv.y;
            dst[(c4 + 2) * 2] = vv.z; dst[(c4 + 3) * 2] = vv.w;
        }
        __syncthreads();

#pragma unroll
        for (int ks = 0; ks < 32; ++ks) {
            int k  = half * 128 + ks * 4;
            int kk = k + 2 * hi;
            v2f a  = *(const v2f*)&S[lm * SST + kc0 + kk];
            v2f bb = Vp[(kk >> 1) * 80 + dbase + lm];
            acc = WMMA_F32(a, bb, acc);
        }
        __syncthreads();
    }

    if (half == 1) {
#pragma unroll
        for (int r = 0; r < 8; ++r) Ored[dt * 256 + r * 32 + lane] = acc[r];
    }
    __syncthreads();
    if (half == 0) {
#pragma unroll
        for (int r = 0; r < 8; ++r) {
            float o = acc[r] + Ored[dt * 256 + r * 32 + lane];
            int qrow = q0 + r + 8 * hi;
            if (qrow < 600)
                ctx[(size_t)(b * 600 + qrow) * 512 + h * 64 + dbase + lm] = o;
        }
    }
}

// ---------------------------------------------------------------------------
extern "C" void kernel_launch(void* const* d_in, const int* in_sizes, int n_in,
                              void* d_out, int out_size, void* d_ws, size_t ws_size,
                              hipStream_t stream)
{
    (void)in_sizes; (void)n_in; (void)out_size; (void)ws_size;

    const float* nodes  = (const float*)d_in[0];
    const float* images = (const float*)d_in[1];
    const float* rel    = (const float*)d_in[2];
    const float* q_w    = (const float*)d_in[3];
    const float* q_b    = (const float*)d_in[4];
    const float* kv_w   = (const float*)d_in[5];
    const float* kv_b   = (const float*)d_in[6];
    const float* cpb_w1 = (const float*)d_in[7];
    const float* cpb_b1 = (const float*)d_in[8];
    const float* cpb_w2 = (const float*)d_in[9];
    const float* cpb_b2 = (const float*)d_in[10];
    const float* out_w  = (const float*)d_in[11];
    const float* out_b  = (const float*)d_in[12];
    float* out = (float*)d_out;

    float* ws   = (float*)d_ws;
    float* qws  = ws;                              // 2400*512
    float* kvws = qws + 2400 * 512;                // 4096*1024
    float* ctx  = kvws + 4096 * 1024;              // 2400*512

    dim3 blk(256);
    gemm_bias_kernel<<<dim3(38, 8),  blk, 0, stream>>>(nodes,  q_w,  q_b,  qws,  2400, 512,  512);
    gemm_bias_kernel<<<dim3(64, 16), blk, 0, stream>>>(images, kv_w, kv_b, kvws, 4096, 1024, 512);

    size_t smem = (size_t)(16 * SST + KVSTAGE + 256 + 128 + 128 + 1024) * sizeof(float);
    attn_kernel<<<dim3(38, 8, 4), blk, smem, stream>>>(qws, kvws, rel,
                                                       cpb_w1, cpb_b1, cpb_w2, cpb_b2, ctx);

    gemm_bias_kernel<<<dim3(38, 8), blk, 0, stream>>>(ctx, out_w, out_b, out, 2400, 512, 512);
}